// NURBSSurface_15221364097068
// MI455X (gfx1250) — compile-verified
//
#include <hip/hip_runtime.h>

#ifndef __has_builtin
#define __has_builtin(x) 0
#endif

// CDNA5 async global->LDS path (gfx1250). Guarded so the host pass and any
// toolchain lacking these builtins still compiles via the fallback.
#if defined(__AMDGCN__) && \
    __has_builtin(__builtin_amdgcn_global_load_async_to_lds_b128) && \
    __has_builtin(__builtin_amdgcn_global_load_async_to_lds_b32) && \
    __has_builtin(__builtin_amdgcn_s_wait_asynccnt)
#define HAVE_ASYNC 1
typedef int v4i __attribute__((vector_size(16)));
typedef __attribute__((address_space(1))) v4i* gptr_b128;
typedef __attribute__((address_space(3))) v4i* lptr_b128;
typedef __attribute__((address_space(1))) int* gptr_b32;
typedef __attribute__((address_space(3))) int* lptr_b32;
#else
#define HAVE_ASYNC 0
#endif

#define OUT_XN 1024
#define OUT_YN 1024
#define NCTRL  256
#define DEG    3
#define KN     (NCTRL + DEG + 1)   // 260 knots per axis
#define EPSV   1e-5f

__device__ __forceinline__ int mod_kn(int x) {
    x %= KN;
    return (x < 0) ? x + KN : x;
}

// ---------------------------------------------------------------------------
// Pass 1: knot normalization.  k = where(k<0, 1e-4, k); k = cumsum(k);
//         k = (k - k[0]) / (k[-1] - k[0]).  260 elems -> serial, trivial.
// block 0 -> x knots, block 1 -> y knots.
// ---------------------------------------------------------------------------
__global__ void knots_kernel(const float* __restrict__ kraw_x,
                             const float* __restrict__ kraw_y,
                             float* __restrict__ kx,
                             float* __restrict__ ky) {
    if (threadIdx.x != 0) return;
    const float* src = (blockIdx.x == 0) ? kraw_x : kraw_y;
    float*       dst = (blockIdx.x == 0) ? kx : ky;
    float c = 0.0f, c0 = 0.0f;
    for (int i = 0; i < KN; ++i) {
        float v = src[i];
        v = (v < 0.0f) ? 1e-4f : v;
        c += v;
        dst[i] = c;
        if (i == 0) c0 = c;
    }
    const float den = c - c0;   // k[-1] - k[0]
    for (int i = 0; i < KN; ++i)
        dst[i] = (dst[i] - c0) / den;
}

// ---------------------------------------------------------------------------
// Pass 2: spans + cubic Cox-de-Boor basis per sample.
// blocks 0..3 -> x samples (1024), blocks 4..7 -> y samples (1024).
// ---------------------------------------------------------------------------
__global__ void __launch_bounds__(256)
basis_kernel(const float* __restrict__ kx, const float* __restrict__ ky,
             int* __restrict__ sx, int* __restrict__ sy,
             float* __restrict__ bx, float* __restrict__ by) {
    const int which = blockIdx.x >> 2;                      // 0 = x, 1 = y
    const int i     = ((blockIdx.x & 3) << 8) + threadIdx.x; // 0..1023
    const float* knots = which ? ky : kx;
    int*   sp = which ? sy : sx;
    float* bb = which ? by : bx;

    // u = linspace(EPS, 1-EPS, 1024)[i]
    const float u = EPSV + (1.0f - 2.0f * EPSV) * ((float)i / 1023.0f);

    // searchsorted(knots, u, side='right') - 1
    int lo = 0, hi = KN;
    while (lo < hi) {
        int mid = (lo + hi) >> 1;
        if (knots[mid] <= u) lo = mid + 1; else hi = mid;
    }
    int span = lo - 1;
    if (u == knots[NCTRL]) span = NCTRL - 1;

    // Cox-de-Boor, degree 3, with the reference's mod-260 knot wrap.
    float left[DEG + 1], right[DEG + 1], cols[DEG + 1];
    cols[0] = 1.0f;
    for (int j = 1; j <= DEG; ++j) {
        left[j]  = u - knots[mod_kn(span + 1 - j)];
        right[j] = knots[mod_kn(span + j)] - u;
        float saved = 0.0f;
        for (int r = 0; r < j; ++r) {
            const float temp = cols[r] / (right[r + 1] + left[j - r]);
            cols[r] = saved + right[r + 1] * temp;
            saved   = left[j - r] * temp;
        }
        cols[j] = saved;
    }

    sp[i] = span;
#pragma unroll
    for (int l = 0; l <= DEG; ++l) bb[i * 4 + l] = cols[l];
}

// ---------------------------------------------------------------------------
// Pass 3: surface evaluation.  One 16x16 pixel tile per 256-thread block.
// Basis rows + spans for the tile are staged into LDS with CDNA5 async
// global->LDS loads (b128/b32) and s_wait_asynccnt.
// ---------------------------------------------------------------------------
__global__ void __launch_bounds__(256)
eval_kernel(const float* __restrict__ cp,
            const int* __restrict__ sx, const int* __restrict__ sy,
            const float* __restrict__ bx, const float* __restrict__ by,
            float* __restrict__ out) {
    __shared__ float s_bx[16 * 4];
    __shared__ float s_by[16 * 4];
    __shared__ int   s_sx[16];
    __shared__ int   s_sy[16];

    const int tid = threadIdx.x;
    const int e0  = (int)blockIdx.y << 4;
    const int f0  = (int)blockIdx.x << 4;

#if HAVE_ASYNC
    if (tid < 16) {
        __builtin_amdgcn_global_load_async_to_lds_b128(
            (gptr_b128)(bx + (size_t)(e0 + tid) * 4),
            (lptr_b128)&s_bx[tid * 4], 0, 0);
    } else if (tid < 32) {
        const int t = tid - 16;
        __builtin_amdgcn_global_load_async_to_lds_b128(
            (gptr_b128)(by + (size_t)(f0 + t) * 4),
            (lptr_b128)&s_by[t * 4], 0, 0);
    } else if (tid < 48) {
        const int t = tid - 32;
        __builtin_amdgcn_global_load_async_to_lds_b32(
            (gptr_b32)(sx + e0 + t),
            (lptr_b32)&s_sx[t], 0, 0);
    } else if (tid < 64) {
        const int t = tid - 48;
        __builtin_amdgcn_global_load_async_to_lds_b32(
            (gptr_b32)(sy + f0 + t),
            (lptr_b32)&s_sy[t], 0, 0);
    }
    __builtin_amdgcn_s_wait_asynccnt(0);
    __syncthreads();
#else
    if (tid < 16) {
#pragma unroll
        for (int l = 0; l < 4; ++l) s_bx[tid * 4 + l] = bx[(size_t)(e0 + tid) * 4 + l];
        s_sx[tid] = sx[e0 + tid];
    } else if (tid < 32) {
        const int t = tid - 16;
#pragma unroll
        for (int l = 0; l < 4; ++l) s_by[t * 4 + l] = by[(size_t)(f0 + t) * 4 + l];
        s_sy[t] = sy[f0 + t];
    }
    __syncthreads();
#endif

    const int le = tid >> 4;      // local e (0..15)
    const int lf = tid & 15;      // local f (0..15)
    const int e  = e0 + le;
    const int f  = f0 + lf;

    const int se = s_sx[le];
    const int sf = s_sy[lf];

    float wx[4], wy[4];
    int   ixr[4], iyr[4];
#pragma unroll
    for (int l = 0; l < 4; ++l) {
        wx[l]  = s_bx[le * 4 + l];
        wy[l]  = s_by[lf * 4 + l];
        ixr[l] = (se - DEG + l + NCTRL) & (NCTRL - 1);   // mod 256
        iyr[l] = (sf - DEG + l + NCTRL) & (NCTRL - 1);
    }

    float ax = 0.0f, ay = 0.0f, az = 0.0f;
#pragma unroll
    for (int l = 0; l < 4; ++l) {
        const float* row = cp + (size_t)ixr[l] * (NCTRL * 3);
        const float  wl  = wx[l];
#pragma unroll
        for (int r = 0; r < 4; ++r) {
            const float  w = wl * wy[r];
            const float* p = row + iyr[r] * 3;
            ax = fmaf(w, p[0], ax);
            ay = fmaf(w, p[1], ay);
            az = fmaf(w, p[2], az);
        }
    }

    const size_t o = ((size_t)e * OUT_YN + f) * 3;
    out[o + 0] = ax;
    out[o + 1] = ay;
    out[o + 2] = az;
}

// ---------------------------------------------------------------------------
// Host launcher
// ---------------------------------------------------------------------------
extern "C" void kernel_launch(void* const* d_in, const int* in_sizes, int n_in,
                              void* d_out, int out_size, void* d_ws, size_t ws_size,
                              hipStream_t stream) {
    const float* cp   = (const float*)d_in[0];   // (256,256,3) f32
    const float* kvx  = (const float*)d_in[1];   // (1,260) f32
    const float* kvy  = (const float*)d_in[2];   // (1,260) f32
    float*       out  = (float*)d_out;           // (1,1024,1024,3) f32

    float* ws = (float*)d_ws;
    float* kx = ws;                 // 272 floats (260 used, padded)
    float* ky = ws + 272;           // 272 floats
    int*   sx = (int*)(ws + 544);   // 1024 ints
    int*   sy = sx + 1024;          // 1024 ints
    float* bx = ws + 544 + 2048;    // 1024*4 floats
    float* by = bx + 4096;          // 1024*4 floats
    // total ~42 KB of workspace

    knots_kernel<<<2, 32, 0, stream>>>(kvx, kvy, kx, ky);
    basis_kernel<<<8, 256, 0, stream>>>(kx, ky, sx, sy, bx, by);
    eval_kernel<<<dim3(OUT_YN / 16, OUT_XN / 16), 256, 0, stream>>>(
        cp, sx, sy, bx, by, out);
}